// SequentialParsingModel_66864050864296
// MI455X (gfx1250) — compile-verified
//
#include <hip/hip_runtime.h>
#include <math.h>

typedef __attribute__((ext_vector_type(2))) float v2f;
typedef __attribute__((ext_vector_type(8))) float v8f;

#define BB 8
#define SS 128
#define DIN 431
#define DPAD 432
#define HD 256
#define G4 1024      // 4*H
#define OUTD 512
#define NEGV -10000.0f
#define LOG2E 1.4426950408889634f

// --- fast transcendental helpers (native CDNA5 TRANS ops when available) ---
__device__ __forceinline__ float fast_tanh(float x) {
#if __has_builtin(__builtin_amdgcn_tanhf)
    return __builtin_amdgcn_tanhf(x);
#else
    return tanhf(x);
#endif
}

__device__ __forceinline__ float fast_sigmoid(float x) {
#if __has_builtin(__builtin_amdgcn_exp2f)
    float e = __builtin_amdgcn_exp2f(-x * LOG2E);   // exp(-x) via v_exp_f32
#else
    float e = expf(-x);
#endif
    return 1.0f / (1.0f + e);
}

// ---------------------------------------------------------------------------
// 0) zero the loss accumulator
__global__ void zero_loss_kernel(float* loss) {
    if (threadIdx.x == 0 && blockIdx.x == 0) loss[0] = 0.0f;
}

// ---------------------------------------------------------------------------
// 1) embedding gather + concat, padded to DPAD columns (col 431 = 0)
__global__ void embed_concat_kernel(const int* __restrict__ words,
                                    const int* __restrict__ pos,
                                    const float* __restrict__ morph,
                                    const float* __restrict__ labels,
                                    const float* __restrict__ word_emb,
                                    const float* __restrict__ pos_emb,
                                    float* __restrict__ sent) {
    int idx = blockIdx.x * blockDim.x + threadIdx.x;
    if (idx >= BB * SS * DPAD) return;
    int r = idx / DPAD;
    int c = idx % DPAD;
    float v;
    if (c < 300)       v = word_emb[words[r] * 300 + c];
    else if (c < 332)  v = pos_emb[pos[r] * 32 + (c - 300)];
    else if (c < 388)  v = morph[r * 56 + (c - 332)];
    else if (c < 431)  v = labels[r * 43 + (c - 388)];
    else               v = 0.0f;
    sent[idx] = v;
}

// ---------------------------------------------------------------------------
// 2) pad Wi (431 x 1024) -> (432 x 1024), last row zero, both directions
__global__ void pad_wi_kernel(const float* __restrict__ wi_f,
                              const float* __restrict__ wi_b,
                              float* __restrict__ pf, float* __restrict__ pb) {
    int idx = blockIdx.x * blockDim.x + threadIdx.x;
    if (idx >= 2 * DPAD * G4) return;
    int dir = idx / (DPAD * G4);
    int e = idx % (DPAD * G4);
    int k = e / G4, n = e % G4;
    float v = 0.0f;
    if (k < DIN) v = dir ? wi_b[k * G4 + n] : wi_f[k * G4 + n];
    if (dir) pb[e] = v; else pf[e] = v;
}

// ---------------------------------------------------------------------------
// 3) generic fp32 WMMA GEMM:  C[M,N] = A[M,K] @ B[K,N] + bias[N]
//    one 16x16 tile per wave, K stepped by 4 via V_WMMA_F32_16X16X4_F32.
__global__ void gemm_wmma_f32_kernel(const float* __restrict__ A,
                                     const float* __restrict__ Bm,
                                     const float* __restrict__ bias,
                                     float* __restrict__ C,
                                     int M, int N, int K) {
    int lane = threadIdx.x & 31;
    int wid = (blockIdx.x * blockDim.x + threadIdx.x) >> 5;
    int ntile = N >> 4;
    int tiles = (M >> 4) * ntile;
    if (wid >= tiles) return;
    int tm = wid / ntile;
    int tn = wid - tm * ntile;

    int col   = tn * 16 + (lane & 15);
    int arow  = tm * 16 + (lane & 15);
    int khalf = (lane >> 4) << 1;

    v8f acc = {0.f, 0.f, 0.f, 0.f, 0.f, 0.f, 0.f, 0.f};
    for (int k = 0; k < K; k += 4) {
        int kb = k + khalf;
        v2f a; a.x = A[arow * K + kb];  a.y = A[arow * K + kb + 1];
        v2f b; b.x = Bm[kb * N + col];  b.y = Bm[(kb + 1) * N + col];
        acc = __builtin_amdgcn_wmma_f32_16x16x4_f32(
            false, a, false, b, (short)0, acc, false, false);
    }
    float bv = bias[col];
    int rbase = tm * 16 + ((lane >> 4) << 3);
    #pragma unroll
    for (int r = 0; r < 8; ++r)
        C[(rbase + r) * N + col] = acc[r] + bv;
}

// ---------------------------------------------------------------------------
// 4) BiLSTM: one block per direction, 1024 threads = 32 waves.
//    h,c live in LDS; per step: g = xproj[:,t,:] + h @ Wh via WMMA
//    (64 N-tiles of 16, 2 per wave), then fused gate update.
__global__ void __launch_bounds__(1024)
lstm_kernel(const float* __restrict__ xproj_f, const float* __restrict__ xproj_b,
            const float* __restrict__ Wh_f,    const float* __restrict__ Wh_b,
            float* __restrict__ enc) {
    __shared__ float h_s[BB * HD];   // 8KB
    __shared__ float c_s[BB * HD];   // 8KB
    __shared__ float g_s[BB * G4];   // 32KB

    int dir = blockIdx.x;
    const float* xproj = dir ? xproj_b : xproj_f;
    const float* Wh    = dir ? Wh_b    : Wh_f;

    int tid  = threadIdx.x;
    int lane = tid & 31;
    int wave = tid >> 5;

    for (int e = tid; e < BB * HD; e += 1024) { h_s[e] = 0.f; c_s[e] = 0.f; }
    __syncthreads();

    int mrow  = lane & 15;
    int mb    = (lane >> 4) << 3;      // 0 or 8
    int khalf = (lane >> 4) << 1;      // 0 or 2

    for (int step = 0; step < SS; ++step) {
        int t = dir ? (SS - 1 - step) : step;

        // --- g = xproj_t + h @ Wh  (M=8 padded to 16, N=1024, K=256)
        for (int tile = wave; tile < 64; tile += 32) {
            int nn = tile * 16 + mrow;
            v8f acc;
            #pragma unroll
            for (int r = 0; r < 8; ++r) {
                int m = mb + r;
                acc[r] = (m < BB) ? xproj[(m * SS + t) * G4 + nn] : 0.f;
            }
            for (int k = 0; k < HD; k += 4) {
                int kb = k + khalf;
                v2f a;
                a.x = (mrow < BB) ? h_s[mrow * HD + kb]     : 0.f;
                a.y = (mrow < BB) ? h_s[mrow * HD + kb + 1] : 0.f;
                v2f b;
                b.x = Wh[kb * G4 + nn];
                b.y = Wh[(kb + 1) * G4 + nn];
                acc = __builtin_amdgcn_wmma_f32_16x16x4_f32(
                    false, a, false, b, (short)0, acc, false, false);
            }
            #pragma unroll
            for (int r = 0; r < 8; ++r) {
                int m = mb + r;
                if (m < BB) g_s[m * G4 + nn] = acc[r];
            }
        }
        __syncthreads();

        // --- fused gate update, 2048 (b,j) elements (serial-path latency!)
        for (int e = tid; e < BB * HD; e += 1024) {
            int b = e >> 8, j = e & 255;
            float ig = g_s[b * G4 + j];
            float fg = g_s[b * G4 + HD + j];
            float gg = g_s[b * G4 + 2 * HD + j];
            float og = g_s[b * G4 + 3 * HD + j];
            float cn = fast_sigmoid(fg) * c_s[e] + fast_sigmoid(ig) * fast_tanh(gg);
            float hn = fast_sigmoid(og) * fast_tanh(cn);
            c_s[e] = cn;
            h_s[e] = hn;
            enc[(b * SS + t) * OUTD + dir * HD + j] = hn;
        }
        __syncthreads();
    }
}

// ---------------------------------------------------------------------------
// 5) fused biaffine scorer + masks + log-softmax + CE + probs
//    one block per (b, i) with i in [1, S); 256 threads = 8 waves.
//    Each lane owns a contiguous 16-float chunk of d -> b128 loads;
//    67M tanh dominate -> native v_tanh_f32.
__global__ void __launch_bounds__(256)
scores_kernel(const float* __restrict__ u, const float* __restrict__ w,
              const float* __restrict__ v, const int* __restrict__ words,
              const int* __restrict__ heads,
              float* __restrict__ probs, float* __restrict__ loss) {
    __shared__ float wrow[OUTD];
    __shared__ float vv[OUTD];
    __shared__ float sc_s[SS];
    __shared__ float red[SS];

    int b = blockIdx.x / (SS - 1);
    int i = blockIdx.x % (SS - 1) + 1;
    int tid = threadIdx.x;
    int lane = tid & 31;
    int wave = tid >> 5;

    for (int d = tid; d < OUTD; d += 256) {
        wrow[d] = w[(b * SS + i) * OUTD + d];
        vv[d]   = v[d];
    }
    __syncthreads();

    // each wave handles columns j = wave, wave+8, ...; lane owns d-chunk [16*lane,16*lane+16)
    int dbase = lane * 16;
    const float4* wr4 = (const float4*)(&wrow[dbase]);
    const float4* vv4 = (const float4*)(&vv[dbase]);

    for (int j = wave; j < SS; j += 8) {
        const float4* ur4 = (const float4*)(u + (b * SS + j) * OUTD + dbase);
        if (j + 8 < SS)
            __builtin_prefetch(u + (b * SS + j + 8) * OUTD + dbase, 0, 1);
        float acc = 0.f;
        #pragma unroll
        for (int q = 0; q < 4; ++q) {
            float4 uu = ur4[q];
            float4 ww = wr4[q];
            float4 vq = vv4[q];
            acc += vq.x * fast_tanh(uu.x + ww.x);
            acc += vq.y * fast_tanh(uu.y + ww.y);
            acc += vq.z * fast_tanh(uu.z + ww.z);
            acc += vq.w * fast_tanh(uu.w + ww.w);
        }
        #pragma unroll
        for (int off = 16; off > 0; off >>= 1)
            acc += __shfl_down(acc, off, 32);
        if (lane == 0) {
            float sc = acc;
            if (j == i) sc = NEGV;                 // enc-row equality == diagonal
            if (words[b * SS + j] == 0) sc = NEGV; // column mask
            sc_s[j] = sc;
        }
    }
    __syncthreads();

    // probs = exp(raw masked logits)
    if (tid < SS)
        probs[(b * (SS - 1) + (i - 1)) * SS + tid] = expf(sc_s[tid]);

    // log-sum-exp over the 128 columns
    if (tid < SS) red[tid] = sc_s[tid];
    __syncthreads();
    for (int s = SS / 2; s > 0; s >>= 1) {
        if (tid < s) red[tid] = fmaxf(red[tid], red[tid + s]);
        __syncthreads();
    }
    float m = red[0];
    __syncthreads();
    if (tid < SS) red[tid] = expf(sc_s[tid] - m);
    __syncthreads();
    for (int s = SS / 2; s > 0; s >>= 1) {
        if (tid < s) red[tid] += red[tid + s];
        __syncthreads();
    }
    if (tid == 0) {
        float lse = m + logf(red[0]);
        int tgt = heads[b * SS + i];
        float ce = -(sc_s[tgt] - lse);
        atomicAdd(loss, ce * (1.0f / (float)BB));
    }
}

// ---------------------------------------------------------------------------
extern "C" void kernel_launch(void* const* d_in, const int* in_sizes, int n_in,
                              void* d_out, int out_size, void* d_ws, size_t ws_size,
                              hipStream_t stream) {
    const int*   words    = (const int*)  d_in[0];
    const int*   pos      = (const int*)  d_in[1];
    const float* morph    = (const float*)d_in[2];
    const float* labels   = (const float*)d_in[3];
    const int*   heads    = (const int*)  d_in[4];
    // d_in[5] sent_ids unused
    const float* word_emb = (const float*)d_in[6];
    const float* pos_emb  = (const float*)d_in[7];
    const float* Wi_f     = (const float*)d_in[8];
    const float* Wh_f     = (const float*)d_in[9];
    const float* b_f      = (const float*)d_in[10];
    const float* Wi_b     = (const float*)d_in[11];
    const float* Wh_b     = (const float*)d_in[12];
    const float* b_b      = (const float*)d_in[13];
    const float* Wu       = (const float*)d_in[14];
    const float* bu       = (const float*)d_in[15];
    const float* Ww       = (const float*)d_in[16];
    const float* bw       = (const float*)d_in[17];
    const float* vvec     = (const float*)d_in[18];

    float* loss  = (float*)d_out;
    float* probs = (float*)d_out + 1;

    // workspace layout (floats)
    float* ws   = (float*)d_ws;
    float* sent = ws;                       // 1024*432
    float* wif  = sent + BB * SS * DPAD;    // 432*1024
    float* wib  = wif  + DPAD * G4;
    float* xf   = wib  + DPAD * G4;         // 1024*1024
    float* xb   = xf   + BB * SS * G4;
    float* enc  = xb   + BB * SS * G4;      // 1024*512
    float* ubuf = enc  + BB * SS * OUTD;
    float* wbuf = ubuf + BB * SS * OUTD;

    zero_loss_kernel<<<1, 1, 0, stream>>>(loss);

    {
        int n = BB * SS * DPAD;
        embed_concat_kernel<<<(n + 255) / 256, 256, 0, stream>>>(
            words, pos, morph, labels, word_emb, pos_emb, sent);
    }
    {
        int n = 2 * DPAD * G4;
        pad_wi_kernel<<<(n + 255) / 256, 256, 0, stream>>>(Wi_f, Wi_b, wif, wib);
    }
    {   // xproj_{f,b} = sent_pad @ Wi_pad + b   (M=1024, N=1024, K=432)
        int tiles = (BB * SS / 16) * (G4 / 16);          // 4096
        int blocks = (tiles + 7) / 8;                    // 8 waves per block
        gemm_wmma_f32_kernel<<<blocks, 256, 0, stream>>>(
            sent, wif, b_f, xf, BB * SS, G4, DPAD);
        gemm_wmma_f32_kernel<<<blocks, 256, 0, stream>>>(
            sent, wib, b_b, xb, BB * SS, G4, DPAD);
    }

    lstm_kernel<<<2, 1024, 0, stream>>>(xf, xb, Wh_f, Wh_b, enc);

    {   // u = enc@Wu + bu ; w = enc@Ww + bw  (M=1024, N=512, K=512)
        int tiles = (BB * SS / 16) * (OUTD / 16);        // 2048
        int blocks = (tiles + 7) / 8;
        gemm_wmma_f32_kernel<<<blocks, 256, 0, stream>>>(
            enc, Wu, bu, ubuf, BB * SS, OUTD, OUTD);
        gemm_wmma_f32_kernel<<<blocks, 256, 0, stream>>>(
            enc, Ww, bw, wbuf, BB * SS, OUTD, OUTD);
    }

    scores_kernel<<<BB * (SS - 1), 256, 0, stream>>>(
        ubuf, wbuf, vvec, words, heads, probs, loss);
}